// ReZSL_14422500180286
// MI455X (gfx1250) — compile-verified
//
#include <hip/hip_runtime.h>
#include <cstdint>
#include <cstddef>

// ReZSL weight-matrix update for MI455X (gfx1250, wave32).
// Memory-bound scatter/reduction (~350 MB traffic -> ~15us at 23.3 TB/s).
// CDNA5 paths used:
//   * global_load_async_to_lds_b128 + s_wait_asynccnt: one HBM read per batch
//     row DMA'd into LDS (bypassing VGPRs), read back once from LDS.
//   * v_wmma_f32_16x16x4_f32: exact-f32 wave-level sum reduction for the L2
//     norms (A = partials, B = ones => row sums), replacing an 8-barrier LDS
//     tree with 2 WMMAs + 8 adds and a single barrier.
//   * L2-resident global_atomic_add_f32 scatter (33 MB accumulator << 192 MB L2).

#define C_NUM 8192
#define D_DIM 1024
#define B_NUM 16384
#define NORM_EPS 1e-10f

typedef float v2f __attribute__((ext_vector_type(2)));
typedef float v8f __attribute__((ext_vector_type(8)));

// Low 32 bits of a flat pointer to LDS == workgroup-relative LDS byte offset
// (flat-shared = {aperture_hi32, lds_off32}; async VDST VGPR wants the offset).
__device__ __forceinline__ unsigned lds_off_u32(const void* p) {
  return (unsigned)(uintptr_t)p;
}

// Wave32 sum reduction on the matrix pipe. A-matrix 16x4 f32 layout:
// VGPR0 = {lanes 0-15: K0, lanes 16-31: K2}, VGPR1 = {K1, K3}. With
// A = (x, 0) and B = ones: D[m,n] = x[m] + x[m+16] (halves folded).
// Sum of the 8 D VGPRs per lane = subtotal of one 16-lane half (rows 0-7 for
// lanes 0-15, rows 8-15 for lanes 16-31). A second identical WMMA folds the
// halves: every lane of D2[0] holds the full 32-lane sum. Exact f32 adds.
__device__ __forceinline__ float wave_sum_wmma(float x) {
  v2f ones;
  ones[0] = 1.0f;
  ones[1] = 1.0f;
  v2f a;
  a[0] = x;
  a[1] = 0.0f;
  v8f c = {};
  v8f d1 = __builtin_amdgcn_wmma_f32_16x16x4_f32(
      /*neg_a=*/false, a, /*neg_b=*/false, ones,
      /*c_mod=*/(short)0, c, /*reuse_a=*/false, /*reuse_b=*/false);
  float s = d1[0] + d1[1] + d1[2] + d1[3] + d1[4] + d1[5] + d1[6] + d1[7];
  v2f a2;
  a2[0] = s;
  a2[1] = 0.0f;
  v8f d2 = __builtin_amdgcn_wmma_f32_16x16x4_f32(
      false, a2, false, ones, (short)0, c, false, false);
  return d2[0];
}

// ---------------------------------------------------------------------------
// K0: zero mean accumulator + counts, featmin = +inf bits
// ---------------------------------------------------------------------------
__global__ __launch_bounds__(256) void rz_init(float4* __restrict__ sums4,
                                               float* __restrict__ counts,
                                               unsigned* __restrict__ featmin) {
  const int i = blockIdx.x * 256 + threadIdx.x;
  if (i < (C_NUM * D_DIM / 4)) sums4[i] = make_float4(0.f, 0.f, 0.f, 0.f);
  if (i < C_NUM) counts[i] = 0.f;
  if (i < D_DIM) featmin[i] = 0x7F800000u;  // +inf
}

// ---------------------------------------------------------------------------
// K1: per-row normalize + squared diff + scatter-add by label.
// One block (256 thr = 8 waves) per batch row; each thread owns 4 elements.
// Rows are DMA'd to LDS with global_load_async_to_lds_b128; norms reduced on
// the WMMA pipe.
// ---------------------------------------------------------------------------
__global__ __launch_bounds__(256) void rz_scatter(const float* __restrict__ pred,
                                                  const float* __restrict__ truth,
                                                  const int* __restrict__ label,
                                                  float* __restrict__ sums,
                                                  float* __restrict__ counts) {
  __shared__ __align__(16) float lp[D_DIM];
  __shared__ __align__(16) float lt[D_DIM];
  __shared__ float red[16];

  const int row = blockIdx.x;
  const int t = threadIdx.x;
  // byte offset of this thread's 16B chunk within the row
  const unsigned goff = (unsigned)row * (unsigned)(D_DIM * 4) + (unsigned)t * 16u;
  const unsigned lpo = lds_off_u32(&lp[t * 4]);
  const unsigned lto = lds_off_u32(&lt[t * 4]);

  // Async DMA: 16B per lane, global(SADDR base + 32b voffset) -> LDS.
  asm volatile("global_load_async_to_lds_b128 %0, %1, %2"
               :: "v"(lpo), "v"(goff), "s"(pred) : "memory");
  asm volatile("global_load_async_to_lds_b128 %0, %1, %2"
               :: "v"(lto), "v"(goff), "s"(truth) : "memory");
  asm volatile("s_wait_asynccnt 0" ::: "memory");  // this wave's DMAs done
  __syncthreads();                                  // all waves' DMAs done

  const float4 p4 = *reinterpret_cast<const float4*>(&lp[t * 4]);
  const float4 t4 = *reinterpret_cast<const float4*>(&lt[t * 4]);

  const float sp = p4.x * p4.x + p4.y * p4.y + p4.z * p4.z + p4.w * p4.w;
  const float st = t4.x * t4.x + t4.y * t4.y + t4.z * t4.z + t4.w * t4.w;

  // Intra-wave reduction on the matrix pipe (EXEC all-1s here).
  const float spw = wave_sum_wmma(sp);
  const float stw = wave_sum_wmma(st);

  const int wid = t >> 5;  // wave32
  if ((t & 31) == 0) {
    red[wid] = spw;
    red[8 + wid] = stw;
  }
  __syncthreads();
  float SP = 0.f, ST = 0.f;
#pragma unroll
  for (int w = 0; w < 8; ++w) {
    SP += red[w];
    ST += red[8 + w];
  }

  const float rp = 1.0f / (sqrtf(SP) + NORM_EPS);
  const float rt = 1.0f / (sqrtf(ST) + NORM_EPS);

  const int lab = label[row];
  float* srow = sums + (size_t)lab * D_DIM + (size_t)t * 4;
  const float dx = p4.x * rp - t4.x * rt;
  const float dy = p4.y * rp - t4.y * rt;
  const float dz = p4.z * rp - t4.z * rt;
  const float dw = p4.w * rp - t4.w * rt;
  atomicAdd(&srow[0], dx * dx);
  atomicAdd(&srow[1], dy * dy);
  atomicAdd(&srow[2], dz * dz);
  atomicAdd(&srow[3], dw * dw);
  if (t == 0) atomicAdd(&counts[lab], 1.0f);
}

// ---------------------------------------------------------------------------
// K2: sums -> mean (in place), per-class masked min -> classmin
// ---------------------------------------------------------------------------
__global__ __launch_bounds__(256) void rz_mean(float* __restrict__ mean,
                                               const float* __restrict__ counts,
                                               float* __restrict__ classmin) {
  __shared__ float rmin[256];
  const int c = blockIdx.x;
  const int t = threadIdx.x;
  const float cnt = counts[c];
  const float inv = cnt > 0.f ? 1.0f / cnt : 0.0f;

  float4* rowp = reinterpret_cast<float4*>(mean + (size_t)c * D_DIM);
  float4 s = rowp[t];
  s.x *= inv; s.y *= inv; s.z *= inv; s.w *= inv;
  rowp[t] = s;

  float m = __builtin_inff();
  if (s.x > 0.f) m = fminf(m, s.x);
  if (s.y > 0.f) m = fminf(m, s.y);
  if (s.z > 0.f) m = fminf(m, s.z);
  if (s.w > 0.f) m = fminf(m, s.w);
  rmin[t] = m;
  __syncthreads();
#pragma unroll
  for (int k = 128; k > 0; k >>= 1) {
    if (t < k) rmin[t] = fminf(rmin[t], rmin[t + k]);
    __syncthreads();
  }
  if (t == 0) classmin[c] = rmin[0];  // +inf if row has no positive entry
}

// ---------------------------------------------------------------------------
// K2b: per-feature masked min over classes. Grid (D/256, 64): each block scans
// 128 classes for 256 features (coalesced) -> 64-way atomic contention only.
// Non-negative floats: uint bit ordering == float ordering.
// ---------------------------------------------------------------------------
__global__ __launch_bounds__(256) void rz_featmin(const float* __restrict__ mean,
                                                  unsigned* __restrict__ featmin) {
  const int f = blockIdx.x * 256 + threadIdx.x;
  const int c0 = blockIdx.y * (C_NUM / 64);
  float m = __builtin_inff();
  for (int c = c0; c < c0 + (C_NUM / 64); ++c) {
    const float v = mean[(size_t)c * D_DIM + f];
    if (v > 0.f) m = fminf(m, v);
  }
  if (m < __builtin_inff()) atomicMin(&featmin[f], __float_as_uint(m));
}

// ---------------------------------------------------------------------------
// K3: weights = w1 * w2, p = p2 = 2 -> (log(v/fmin)+1)^2 * (log(v/cmin)+1)^2
// valid == (v > 0): a positive element implies its row and column have a
// positive entry, so `valid = mask & has_pos` collapses to `mask`.
// ---------------------------------------------------------------------------
__global__ __launch_bounds__(256) void rz_weights(const float* __restrict__ mean,
                                                  const float* __restrict__ classmin,
                                                  const unsigned* __restrict__ featmin,
                                                  float* __restrict__ wout) {
  const int c = blockIdx.x;
  const int t = threadIdx.x;
  const float cm = classmin[c];

  const float4 v4 = reinterpret_cast<const float4*>(mean + (size_t)c * D_DIM)[t];
  const uint4 f4 = reinterpret_cast<const uint4*>(featmin)[t];

  const float v[4] = {v4.x, v4.y, v4.z, v4.w};
  const unsigned fb[4] = {f4.x, f4.y, f4.z, f4.w};
  float w[4];
#pragma unroll
  for (int i = 0; i < 4; ++i) {
    if (v[i] > 0.f) {
      const float fm = __uint_as_float(fb[i]);
      const float a = logf(v[i] / fm) + 1.0f;  // w1 base, p  = 2
      const float b = logf(v[i] / cm) + 1.0f;  // w2 base, p2 = 2
      w[i] = (a * a) * (b * b);
    } else {
      w[i] = 1.0f;
    }
  }
  reinterpret_cast<float4*>(wout + (size_t)c * D_DIM)[t] =
      make_float4(w[0], w[1], w[2], w[3]);
}

// ---------------------------------------------------------------------------
extern "C" void kernel_launch(void* const* d_in, const int* in_sizes, int n_in,
                              void* d_out, int out_size, void* d_ws, size_t ws_size,
                              hipStream_t stream) {
  const float* pred = (const float*)d_in[0];
  const float* truth = (const float*)d_in[1];
  const int* label = (const int*)d_in[2];

  float* mean = (float*)d_out;                        // [C, D] (also sum accum)
  float* wout = mean + (size_t)C_NUM * D_DIM;         // [C, D]

  float* counts = (float*)d_ws;                       // C floats
  unsigned* featmin = (unsigned*)(counts + C_NUM);    // D u32 (float bits)
  float* classmin = (float*)(featmin + D_DIM);        // C floats
  (void)in_sizes; (void)n_in; (void)out_size; (void)ws_size;

  rz_init<<<(C_NUM * D_DIM / 4 + 255) / 256, 256, 0, stream>>>(
      (float4*)mean, counts, featmin);
  rz_scatter<<<B_NUM, 256, 0, stream>>>(pred, truth, label, mean, counts);
  rz_mean<<<C_NUM, 256, 0, stream>>>(mean, counts, classmin);
  rz_featmin<<<dim3(D_DIM / 256, 64), 256, 0, stream>>>(mean, featmin);
  rz_weights<<<C_NUM, 256, 0, stream>>>(mean, classmin, featmin, wout);
}